// DampedLinOSSLayer_77481210020590
// MI455X (gfx1250) — compile-verified
//
#include <hip/hip_runtime.h>
#include <math.h>

// ---------------------------------------------------------------------------
// DampedLinOSS fused kernels for gfx1250 (wave32, fp32 WMMA path)
//   K0: prep  -> pair-deinterleave B and C weights into ws:
//                 B4[p][h/2][4] = [Bre(h), Bre(h+1), Bim(h), Bim(h+1)]
//                 C4[h][p/2][4] = [Cre(p), Cre(p+1), -Cim(p), -Cim(p+1)]
//                so one global_load_b128 yields two even-aligned WMMA B-frags.
//   K1: per (batch, segment) -> B-projection (WMMA) + local scan -> seg-end state
//   K2: combine segment carries with lam^SEGLEN (tiny)
//   K3: B-projection (WMMA) + scan w/ carry + C-projection (WMMA) + D*u epilogue
// ---------------------------------------------------------------------------

typedef __attribute__((ext_vector_type(2))) float v2f;
typedef __attribute__((ext_vector_type(4))) float v4f;
typedef __attribute__((ext_vector_type(8))) float v8f;

#define BATCH   32
#define LENGTH  2048
#define HIDDEN  128
#define SSM     256
#define NSEG    8
#define SEGLEN  (LENGTH / NSEG)   // 256
#define CH      16                // timesteps per chunk (WMMA M)
#define NCHUNK  (SEGLEN / CH)     // 16

#define U_STRIDE 130              // padded LDS stride for u chunk (bank-conflict free)
#define S_STRIDE 258              // padded LDS stride for bu/state chunks

#define BQ (SSM * HIDDEN / 2)     // 16384 quads in B (= C)

__device__ __forceinline__ void project_lam(float Ad, float Gd, float st_raw,
                                            float& lr, float& li) {
    // sigmoid step, relu damping, clamp A into complex-eigenvalue band
    float step  = 1.0f / (1.0f + expf(-st_raw));
    float g     = fmaxf(Gd, 0.0f);
    float denom = fmaxf(step * step, 1e-6f);
    float s     = step * g;
    float base  = sqrtf(fmaxf(1.0f + s, 1e-6f));
    float a_lo  = (2.0f + s - 2.0f * base) / denom;
    float a_hi  = (2.0f + s + 2.0f * base) / denom;
    float a     = a_lo + fmaxf(Ad - a_lo, 0.0f) - fmaxf(Ad - a_hi, 0.0f);
    float S     = 1.0f / (1.0f + step * g);
    float T     = S + 1.0f - step * step * S * a;
    lr = 0.5f * T;
    li = sqrtf(fmaxf(S - 0.25f * T * T, 0.0f));
}

// V_WMMA_F32_16X16X4_F32: D(16x16,f32) = A(16x4,f32) * B(4x16,f32) + C
__device__ __forceinline__ v8f wmma_f32(v2f a, v2f b, v8f c) {
    return __builtin_amdgcn_wmma_f32_16x16x4_f32(
        /*neg_a=*/false, a, /*neg_b=*/false, b,
        /*c_mod=*/(short)0, c, /*reuse_a=*/false, /*reuse_b=*/false);
}

// A-operand fragment (16x4, MxK) from an LDS tile [16][stride]:
//   lane<16:  M=lane,    K = k0,   k0+1
//   lane>=16: M=lane-16, K = k0+2, k0+3
__device__ __forceinline__ v2f load_afrag(const float* lds, int stride, int lane, int k0) {
    int row = lane & 15;
    int col = k0 + ((lane >> 4) << 1);
    const float* p = lds + row * stride + col;
    v2f a; a.x = p[0]; a.y = p[1];
    return a;
}

// Per-lane base pointer into a pair-deinterleaved weight array W4[n][k/2][4].
__device__ __forceinline__ const float* frag_base(const float* __restrict__ W4,
                                                  int kdim, int n0, int lane) {
    int n = lane & 15;
    return W4 + ((size_t)((n0 + n) * (kdim >> 1) + (lane >> 4)) << 2);
}

// One b128 -> both (re, im) B-operand fragments, already adjacent/even-aligned.
struct frag2 { v2f re; v2f im; };
__device__ __forceinline__ frag2 load_frag_pair(const float* __restrict__ base, int k0) {
    v4f f = *(const v4f*)(base + k0 * 2);   // immediate offset in unrolled loop
    frag2 r;
    r.re = __builtin_shufflevector(f, f, 0, 1);
    r.im = __builtin_shufflevector(f, f, 2, 3);
    return r;
}

// K0: pair-deinterleave B and C (C imag negated) into workspace.
__global__ __launch_bounds__(256)
void linoss_prep_kernel(const float* __restrict__ Bg,
                        const float* __restrict__ Cg,
                        float* __restrict__ B4,
                        float* __restrict__ C4) {
    int i = blockIdx.x * 256 + threadIdx.x;   // quad index
    if (i < BQ) {
        v4f f = *(const v4f*)(Bg + (size_t)i * 4);   // [re,im,re',im']
        v4f o; o.x = f.x; o.y = f.z; o.z = f.y; o.w = f.w;
        *(v4f*)(B4 + (size_t)i * 4) = o;
    } else {
        int j = i - BQ;
        v4f f = *(const v4f*)(Cg + (size_t)j * 4);
        v4f o; o.x = f.x; o.y = f.z; o.z = -f.y; o.w = -f.w;
        *(v4f*)(C4 + (size_t)j * 4) = o;
    }
}

// MODE 0: partial pass (compute segment-end local state only)
// MODE 1: full pass (carry-in, states, C-projection, output)
template <int MODE>
__global__ __launch_bounds__(256)
void linoss_seg_kernel(const float* __restrict__ u,
                       const float* __restrict__ A_diag,
                       const float* __restrict__ G_diag,
                       const float* __restrict__ steps,
                       const float* __restrict__ B4,
                       const float* __restrict__ C4,
                       const float* __restrict__ Dv,
                       float* __restrict__ xend_r, float* __restrict__ xend_i,
                       const float* __restrict__ carry_r,
                       const float* __restrict__ carry_i,
                       float* __restrict__ out) {
    __shared__ float u_lds[CH * U_STRIDE];
    __shared__ float bu_r [CH * S_STRIDE];   // bu, then states in-place (MODE 1)
    __shared__ float bu_i [CH * S_STRIDE];

    const int seg  = blockIdx.x;
    const int b    = blockIdx.y;
    const int tid  = threadIdx.x;        // 0..255, also the SSM channel p
    const int lane = tid & 31;
    const int wv   = tid >> 5;           // 0..7

    // Per-channel eigenvalue
    float lr, li;
    project_lam(A_diag[tid], G_diag[tid], steps[tid], lr, li);

    // Recurrence state
    float xr = 0.0f, xi = 0.0f;
    if (MODE == 1) {
        size_t co = ((size_t)b * NSEG + seg) * SSM + tid;
        xr = carry_r[co];
        xi = carry_i[co];
    }

    // GEMM1: wave handles p-tiles {wv, wv+8}, both re and im parts.
    // GEMM2: wave handles h-tile wv.
    const int h0      = wv * 16;
    const float dcoef = Dv[h0 + (lane & 15)];

    // Chunk-invariant per-lane fragment base pointers (k indexed by immediate).
    const float* bfrag0 = frag_base(B4, HIDDEN, (wv    ) * 16, lane);
    const float* bfrag1 = frag_base(B4, HIDDEN, (wv + 8) * 16, lane);
    const float* cfrag  = frag_base(C4, SSM,    h0,            lane);

    const float* ubase = u   + ((size_t)b * LENGTH + (size_t)seg * SEGLEN) * HIDDEN;
    float*       obase = out + ((size_t)b * LENGTH + (size_t)seg * SEGLEN) * HIDDEN;

    const v8f vzero = {};

    for (int c = 0; c < NCHUNK; ++c) {
        const int t0 = c * CH;

        // ---- Phase A: stage u chunk [16][128] into LDS ----
        for (int i = tid; i < CH * (HIDDEN / 4); i += 256) {
            int row  = i >> 5;                 // / (HIDDEN/4)
            int col4 = i & 31;                 // % (HIDDEN/4)
            v4f val = *(const v4f*)(ubase + (size_t)(t0 + row) * HIDDEN + col4 * 4);
            float* dst = &u_lds[row * U_STRIDE + col4 * 4];
            dst[0] = val.x; dst[1] = val.y; dst[2] = val.z; dst[3] = val.w;
        }
        // Prefetch next chunk's u tile (8 KB) while this chunk computes.
        if (c + 1 < NCHUNK) {
            const float* nxt = ubase + (size_t)(t0 + CH) * HIDDEN;
            __builtin_prefetch(nxt + tid * 8, 0, 3);
        }
        __syncthreads();

        // ---- Phase B: bu = u @ B^T  (WMMA fp32, K=HIDDEN, re+im together) ----
        v8f accr[2] = {vzero, vzero};
        v8f acci[2] = {vzero, vzero};
        for (int k0 = 0; k0 < HIDDEN; k0 += 4) {
            v2f af = load_afrag(u_lds, U_STRIDE, lane, k0);   // shared by all tiles
            frag2 b0 = load_frag_pair(bfrag0, k0);
            accr[0] = wmma_f32(af, b0.re, accr[0]);
            acci[0] = wmma_f32(af, b0.im, acci[0]);
            frag2 b1 = load_frag_pair(bfrag1, k0);
            accr[1] = wmma_f32(af, b1.re, accr[1]);
            acci[1] = wmma_f32(af, b1.im, acci[1]);
        }
        {   // scatter D tiles to bu (C/D layout: VGPR r -> M=r / M=r+8)
            int n    = lane & 15;
            int mofs = (lane >> 4) << 3;
#pragma unroll
            for (int i = 0; i < 2; ++i) {
                int p0 = (wv + 8 * i) * 16;
#pragma unroll
                for (int r = 0; r < 8; ++r) {
                    bu_r[(mofs + r) * S_STRIDE + p0 + n] = accr[i][r];
                    bu_i[(mofs + r) * S_STRIDE + p0 + n] = acci[i][r];
                }
            }
        }
        __syncthreads();

        // ---- Phase C: complex diagonal scan over the chunk (thread = channel),
        //      states written in place over bu ----
#pragma unroll
        for (int t = 0; t < CH; ++t) {
            float br  = bu_r[t * S_STRIDE + tid];
            float bi  = bu_i[t * S_STRIDE + tid];
            float nxr = lr * xr - li * xi + br;
            float nxi = lr * xi + li * xr + bi;
            xr = nxr; xi = nxi;
            if (MODE == 1) {
                bu_r[t * S_STRIDE + tid] = xr;
                bu_i[t * S_STRIDE + tid] = xi;
            }
        }
        __syncthreads();

        if (MODE == 1) {
            // ---- Phase D: out = Re(X) @ Cr^T + Im(X) @ (-Ci)^T + D*u ----
            v8f oacc = vzero;
            for (int k0 = 0; k0 < SSM; k0 += 4) {
                frag2 cf = load_frag_pair(cfrag, k0);   // im part pre-negated
                v2f ar = load_afrag(bu_r, S_STRIDE, lane, k0);
                oacc = wmma_f32(ar, cf.re, oacc);
                v2f ai = load_afrag(bu_i, S_STRIDE, lane, k0);
                oacc = wmma_f32(ai, cf.im, oacc);
            }
            {
                int n    = lane & 15;
                int mofs = (lane >> 4) << 3;
#pragma unroll
                for (int r = 0; r < 8; ++r) {
                    int m = mofs + r;
                    float val = oacc[r] + u_lds[m * U_STRIDE + h0 + n] * dcoef;
                    obase[(size_t)(t0 + m) * HIDDEN + h0 + n] = val;
                }
            }
            __syncthreads();   // bu/u reused next chunk
        }
    }

    if (MODE == 0) {
        size_t o = ((size_t)b * NSEG + seg) * SSM + tid;
        xend_r[o] = xr;
        xend_i[o] = xi;
    }
}

// Tiny inter-segment carry combine: carry[s] = lam^SEGLEN * carry[s-1] + xend[s-1]
__global__ __launch_bounds__(256)
void linoss_carry_kernel(const float* __restrict__ A_diag,
                         const float* __restrict__ G_diag,
                         const float* __restrict__ steps,
                         const float* __restrict__ xend_r,
                         const float* __restrict__ xend_i,
                         float* __restrict__ carry_r,
                         float* __restrict__ carry_i) {
    int idx = blockIdx.x * 256 + threadIdx.x;   // (b, p) pair
    int b = idx >> 8;
    int p = idx & 255;
    float lr, li;
    project_lam(A_diag[p], G_diag[p], steps[p], lr, li);
    // lam^SEGLEN via 8 complex squarings (2^8 == 256 == SEGLEN)
    float pr = lr, pi = li;
#pragma unroll
    for (int i = 0; i < 8; ++i) {
        float nr = pr * pr - pi * pi;
        float ni = 2.0f * pr * pi;
        pr = nr; pi = ni;
    }
    float cr = 0.0f, ci = 0.0f;
    for (int s = 0; s < NSEG; ++s) {
        size_t o = ((size_t)b * NSEG + s) * SSM + p;
        carry_r[o] = cr;
        carry_i[o] = ci;
        float er = xend_r[o], ei = xend_i[o];
        float nr = pr * cr - pi * ci + er;
        float ni = pr * ci + pi * cr + ei;
        cr = nr; ci = ni;
    }
}

extern "C" void kernel_launch(void* const* d_in, const int* in_sizes, int n_in,
                              void* d_out, int out_size, void* d_ws, size_t ws_size,
                              hipStream_t stream) {
    const float* u  = (const float*)d_in[0];
    const float* A  = (const float*)d_in[1];
    const float* G  = (const float*)d_in[2];
    const float* st = (const float*)d_in[3];
    const float* B  = (const float*)d_in[4];
    const float* C  = (const float*)d_in[5];
    const float* D  = (const float*)d_in[6];
    float* out = (float*)d_out;
    float* ws  = (float*)d_ws;

    const size_t wq = (size_t)BQ * 4;              // floats per weight array (65536)
    const size_t n  = (size_t)BATCH * NSEG * SSM;  // 65536 carries
    float* B4      = ws;
    float* C4      = ws + wq;
    float* xend_r  = ws + 2 * wq;
    float* xend_i  = ws + 2 * wq + n;
    float* carry_r = ws + 2 * wq + 2 * n;
    float* carry_i = ws + 2 * wq + 3 * n;

    dim3 grid(NSEG, BATCH), blk(256);
    linoss_prep_kernel<<<(2 * BQ) / 256, blk, 0, stream>>>(B, C, B4, C4);
    linoss_seg_kernel<0><<<grid, blk, 0, stream>>>(u, A, G, st, B4, C4, D,
                                                   xend_r, xend_i,
                                                   nullptr, nullptr, nullptr);
    linoss_carry_kernel<<<BATCH, 256, 0, stream>>>(A, G, st, xend_r, xend_i,
                                                   carry_r, carry_i);
    linoss_seg_kernel<1><<<grid, blk, 0, stream>>>(u, A, G, st, B4, C4, D,
                                                   nullptr, nullptr,
                                                   carry_r, carry_i, out);
}